// GCN_58179626991898
// MI455X (gfx1250) — compile-verified
//
#include <hip/hip_runtime.h>
#include <math.h>

typedef float v2f __attribute__((ext_vector_type(2)));
typedef float v8f __attribute__((ext_vector_type(8)));

__device__ __forceinline__ void atomAddF(float* p, float v) {
  __hip_atomic_fetch_add(p, v, __ATOMIC_RELAXED, __HIP_MEMORY_SCOPE_AGENT);
}

// ---------------------------------------------------------------------------
// Pack W[128][128] (row-major, K x N) into WMMA-B-ready float2 pairs:
// Wp[(s*2+half)*128 + n] = { W[(4s+2*half)*128+n], W[(4s+2*half+1)*128+n] }
// so the GEMM inner loop does ONE 8-byte load per B fragment.
// ---------------------------------------------------------------------------
__global__ void gcn_packW(const float* __restrict__ W, float2* __restrict__ Wp) {
  int i = blockIdx.x * blockDim.x + threadIdx.x;      // 0 .. 64*128-1 pairs
  if (i >= 64 * 128) return;
  int n  = i & 127;
  int sh = i >> 7;                                    // sh = s*2 + half
  int k  = 2 * sh;                                    // = 4s + 2*half
  float2 p;
  p.x = W[(k + 0) * 128 + n];
  p.y = W[(k + 1) * 128 + n];
  Wp[i] = p;
}

// ---------------------------------------------------------------------------
// Degree (f32 atomics over dst), then dinv = rsqrt(deg + 1 self-loop) in place
// ---------------------------------------------------------------------------
__global__ void gcn_deg(const long long* __restrict__ dst, float* __restrict__ deg, int E) {
  int i = blockIdx.x * blockDim.x + threadIdx.x;
  if (i < E) atomAddF(&deg[dst[i]], 1.0f);
}

__global__ void gcn_dinv(float* __restrict__ deg, int N) {
  int i = blockIdx.x * blockDim.x + threadIdx.x;
  if (i < N) deg[i] = rsqrtf(deg[i] + 1.0f);
}

// ---------------------------------------------------------------------------
// Y[N x 128] = X[N x 128] @ W[128 x 128] using V_WMMA_F32_16X16X4_F32.
// One wave per 16-row strip; 8 v8f accumulators cover all 128 output cols.
// A (16x4 f32): lanes 0-15 -> M=l, K = k..k+1 ; lanes 16-31 -> M=l, K = k+2..k+3
// B (4x16 f32): VGPR v, lanes 0-15 = row K=k+v ; lanes 16-31 = row K=k+v+2
// C/D: VGPR r, lanes 0-15 = (M=r, N=l) ; lanes 16-31 = (M=r+8, N=l)
// ---------------------------------------------------------------------------
__global__ void gcn_gemm_wmma(const float* __restrict__ X,
                              const float2* __restrict__ Wp,
                              float* __restrict__ Y, int nstrips) {
  int wave  = threadIdx.x >> 5;
  int strip = blockIdx.x * 8 + wave;          // wave-uniform -> EXEC stays full
  if (strip >= nstrips) return;
  int lane = threadIdx.x & 31;
  int half = lane >> 4;                       // 0 or 1
  int l    = lane & 15;
  int m0   = strip * 16;

  const float* arow = X + (size_t)(m0 + l) * 128;

  v8f acc[8];
#pragma unroll
  for (int t = 0; t < 8; ++t) acc[t] = (v8f)0.0f;

#pragma unroll 4
  for (int s = 0; s < 32; ++s) {              // K = 4*s
    float2 af = *(const float2*)(arow + 4 * s + 2 * half);
    v2f a; a.x = af.x; a.y = af.y;
    const float2* brow = Wp + (size_t)(s * 2 + half) * 128;
#pragma unroll
    for (int t = 0; t < 8; ++t) {
      float2 bf = brow[t * 16 + l];
      v2f b; b.x = bf.x; b.y = bf.y;
      acc[t] = __builtin_amdgcn_wmma_f32_16x16x4_f32(
          false, a, false, b, (short)0, acc[t], false, false);
    }
  }

#pragma unroll
  for (int t = 0; t < 8; ++t) {
#pragma unroll
    for (int r = 0; r < 8; ++r) {
      Y[(size_t)(m0 + r + 8 * half) * 128 + t * 16 + l] = acc[t][r];
    }
  }
}

// ---------------------------------------------------------------------------
// Edge scatter: one wave per edge; 32 lanes x float4 = 128 features.
// acc[dst][:] += h[src][:] * dinv[src]*dinv[dst]   (native f32 atomics)
// ---------------------------------------------------------------------------
__global__ void gcn_scatter(const long long* __restrict__ src,
                            const long long* __restrict__ dst,
                            const float* __restrict__ h,
                            const float* __restrict__ dinv,
                            float* __restrict__ acc, int E) {
  int e = blockIdx.x * 8 + (threadIdx.x >> 5);
  if (e >= E) return;
  int lane = threadIdx.x & 31;
  long long s = src[e];
  long long d = dst[e];
  float norm = dinv[s] * dinv[d];
  float4 v = *(const float4*)(h + (size_t)s * 128 + lane * 4);
  float* base = acc + (size_t)d * 128 + lane * 4;
  atomAddF(base + 0, v.x * norm);
  atomAddF(base + 1, v.y * norm);
  atomAddF(base + 2, v.z * norm);
  atomAddF(base + 3, v.w * norm);
}

// ---------------------------------------------------------------------------
// out[i][c] = relu(acc[i][c] + hpre[i][c]*dinv[i]^2 + bias[c])
// (self-loop term folded in here; out may alias hpre elementwise)
// ---------------------------------------------------------------------------
__global__ void gcn_finalize(const float* __restrict__ acc,
                             const float* __restrict__ hpre,
                             float* __restrict__ out,
                             const float* __restrict__ dinv,
                             const float* __restrict__ bias, int N) {
  int i = blockIdx.x * blockDim.x + threadIdx.x;   // over N*32 float4 groups
  if (i >= N * 32) return;
  int row = i >> 5;
  int c0  = (i & 31) * 4;
  float di = dinv[row];
  float sl = di * di;
  size_t base = (size_t)i * 4;
  float4 a = *(const float4*)(acc + base);
  float4 hv = *(const float4*)(hpre + base);
  float4 r;
  r.x = fmaxf(a.x + hv.x * sl + bias[c0 + 0], 0.0f);
  r.y = fmaxf(a.y + hv.y * sl + bias[c0 + 1], 0.0f);
  r.z = fmaxf(a.z + hv.z * sl + bias[c0 + 2], 0.0f);
  r.w = fmaxf(a.w + hv.w * sl + bias[c0 + 3], 0.0f);
  *(float4*)(out + base) = r;
}

// ---------------------------------------------------------------------------
// Fused FC (128 -> 40) + log_softmax. Wfc staged in LDS (20 KB).
// One wave per node; lane holds h[lane*4..+3]; __shfl broadcasts the row.
// Lane l computes class l; lanes 0..7 also compute class 32+l.
// ---------------------------------------------------------------------------
__global__ void gcn_fc_lsm(const float* __restrict__ h,
                           const float* __restrict__ Wfc,
                           const float* __restrict__ bfc,
                           float* __restrict__ out, int N) {
  __shared__ float Wl[128 * 40];
  __shared__ float bl[40];
  for (int i = threadIdx.x; i < 128 * 40; i += blockDim.x) Wl[i] = Wfc[i];
  if (threadIdx.x < 40) bl[threadIdx.x] = bfc[threadIdx.x];
  __syncthreads();

  int lane = threadIdx.x & 31;
  int wave = threadIdx.x >> 5;
  int wavesTotal = gridDim.x * (blockDim.x >> 5);
  bool v2ok = lane < 8;
  int c1 = lane;
  int c2 = v2ok ? (32 + lane) : 39;

  for (int node = blockIdx.x * (blockDim.x >> 5) + wave; node < N; node += wavesTotal) {
    float4 hv = *(const float4*)(h + (size_t)node * 128 + lane * 4);
    float d1 = 0.0f, d2 = 0.0f;
#pragma unroll
    for (int k4 = 0; k4 < 32; ++k4) {
      float h0 = __shfl(hv.x, k4, 32);
      float h1 = __shfl(hv.y, k4, 32);
      float h2 = __shfl(hv.z, k4, 32);
      float h3 = __shfl(hv.w, k4, 32);
      int kb = k4 * 4;
      d1 += h0 * Wl[(kb + 0) * 40 + c1] + h1 * Wl[(kb + 1) * 40 + c1]
          + h2 * Wl[(kb + 2) * 40 + c1] + h3 * Wl[(kb + 3) * 40 + c1];
      d2 += h0 * Wl[(kb + 0) * 40 + c2] + h1 * Wl[(kb + 1) * 40 + c2]
          + h2 * Wl[(kb + 2) * 40 + c2] + h3 * Wl[(kb + 3) * 40 + c2];
    }
    d1 += bl[c1];
    d2 += bl[c2];

    float m = fmaxf(d1, v2ok ? d2 : -INFINITY);
#pragma unroll
    for (int off = 16; off; off >>= 1) m = fmaxf(m, __shfl_xor(m, off, 32));
    float sum = expf(d1 - m) + (v2ok ? expf(d2 - m) : 0.0f);
#pragma unroll
    for (int off = 16; off; off >>= 1) sum += __shfl_xor(sum, off, 32);
    float lse = m + logf(sum);

    out[(size_t)node * 40 + c1] = d1 - lse;
    if (v2ok) out[(size_t)node * 40 + (32 + lane)] = d2 - lse;
  }
}

// ---------------------------------------------------------------------------
extern "C" void kernel_launch(void* const* d_in, const int* in_sizes, int n_in,
                              void* d_out, int out_size, void* d_ws, size_t ws_size,
                              hipStream_t stream) {
  const float*     x   = (const float*)d_in[0];
  const float*     W1  = (const float*)d_in[1];
  const float*     b1  = (const float*)d_in[2];
  const float*     W2  = (const float*)d_in[3];
  const float*     b2  = (const float*)d_in[4];
  const float*     Wfc = (const float*)d_in[5];
  const float*     bfc = (const float*)d_in[6];
  const long long* ei  = (const long long*)d_in[7];

  const int N = in_sizes[0] / 128;
  const int E = in_sizes[7] / 2;
  const long long* srcIdx = ei;
  const long long* dstIdx = ei + E;

  // workspace carve-up (256B aligned)
  auto align256 = [](size_t v) { return (v + 255) & ~(size_t)255; };
  size_t szMat = align256((size_t)N * 128 * sizeof(float));
  char* w = (char*)d_ws;
  float*  bufA = (float*)(w);                    // N x 128
  float*  bufB = (float*)(w + szMat);            // N x 128
  float*  dinv = (float*)(w + 2 * szMat);        // N
  size_t  szDinv = align256((size_t)N * sizeof(float));
  float2* Wp1 = (float2*)(w + 2 * szMat + szDinv);           // 64 KB
  float2* Wp2 = (float2*)(w + 2 * szMat + szDinv + 65536);   // 64 KB

  const int nstrips   = N / 16;                        // N = 100000 -> 6250
  const int gemmGrid  = (nstrips + 7) / 8;
  const int scatGrid  = (E + 7) / 8;
  const int finGrid   = (N * 32 + 255) / 256;

  // --- degree / dinv ---
  hipMemsetAsync(dinv, 0, (size_t)N * sizeof(float), stream);
  gcn_deg<<<(E + 255) / 256, 256, 0, stream>>>(dstIdx, dinv, E);
  gcn_dinv<<<(N + 255) / 256, 256, 0, stream>>>(dinv, N);

  // --- pack weights for WMMA B-layout ---
  gcn_packW<<<32, 256, 0, stream>>>(W1, Wp1);
  gcn_packW<<<32, 256, 0, stream>>>(W2, Wp2);

  // --- layer 1: bufA = x@W1 ; bufB = scatter ; bufA = relu(...) ---
  gcn_gemm_wmma<<<gemmGrid, 256, 0, stream>>>(x, Wp1, bufA, nstrips);
  hipMemsetAsync(bufB, 0, (size_t)N * 128 * sizeof(float), stream);
  gcn_scatter<<<scatGrid, 256, 0, stream>>>(srcIdx, dstIdx, bufA, dinv, bufB, E);
  gcn_finalize<<<finGrid, 256, 0, stream>>>(bufB, bufA, bufA, dinv, b1, N);

  // --- layer 2: bufB = bufA@W2 ; bufA = scatter ; bufB = relu(...) ---
  gcn_gemm_wmma<<<gemmGrid, 256, 0, stream>>>(bufA, Wp2, bufB, nstrips);
  hipMemsetAsync(bufA, 0, (size_t)N * 128 * sizeof(float), stream);
  gcn_scatter<<<scatGrid, 256, 0, stream>>>(srcIdx, dstIdx, bufB, dinv, bufA, E);
  gcn_finalize<<<finGrid, 256, 0, stream>>>(bufA, bufB, bufB, dinv, b2, N);

  // --- FC + log_softmax ---
  gcn_fc_lsm<<<2048, 256, 0, stream>>>(bufB, Wfc, bfc, (float*)d_out, N);
}